// PreFeatureExtractor_69922067579633
// MI455X (gfx1250) — compile-verified
//
#include <hip/hip_runtime.h>
#include <math.h>

typedef __attribute__((ext_vector_type(2))) float v2f;
typedef __attribute__((ext_vector_type(8))) float v8f;

#define NPTS   16384
#define KNN    20
#define COUT   32
#define WAVES  4
#define BLOCK  (WAVES * 32)
#define ROWS_PER_WAVE 16
#define ROWS_PER_BLOCK (WAVES * ROWS_PER_WAVE)
#define NTILES (NPTS / 16)
#define FBIG   3.0e38f

__global__ __launch_bounds__(BLOCK) void edgeconv_knn_kernel(
    const float* __restrict__ X,   // [N][3]
    const float* __restrict__ W,   // [32][6]
    const float* __restrict__ Bv,  // [32]
    float* __restrict__ Out)       // [N][32]
{
  // Per-wave scratch for the 16x16 WMMA tile (stride 17 -> conflict-light)
  __shared__ float tileS[WAVES][16][17];
  // Per-row merge area: two sorted 20-lists (lane pair halves)
  __shared__ float mD[WAVES][16][2 * KNN];
  __shared__ int   mI[WAVES][16][2 * KNN];
  __shared__ float Wsh[COUT * 6];
  __shared__ float Bsh[COUT];

  const int tid  = threadIdx.x;
  const int wave = tid >> 5;
  const int lane = tid & 31;
  const int hi   = lane >> 4;   // which half of the wave
  const int col  = lane & 15;

  for (int t = tid; t < COUT * 6; t += BLOCK) Wsh[t] = W[t];
  for (int t = tid; t < COUT;     t += BLOCK) Bsh[t] = Bv[t];
  __syncthreads();

  const int i0  = (blockIdx.x * WAVES + wave) * ROWS_PER_WAVE;
  const int myI = i0 + col;     // row owned by this lane (pair shares a row)

  // ---- A tile (loop-invariant). ISA layout (f32 16x4):
  //  lanes 0-15 : VGPR0=K0, VGPR1=K1 ; lanes 16-31 : VGPR0=K2, VGPR1=K3.
  //  A row = [-2x, -2y, -2z, 1]  so D = A*B = ||xj||^2 - 2 xi.xj  (same
  //  per-row ordering as d^2; the ||xi||^2 constant cannot change top-k).
  const float xix = X[3 * myI + 0];
  const float xiy = X[3 * myI + 1];
  const float xiz = X[3 * myI + 2];
  v2f Areg;
  Areg[0] = hi ? (-2.0f * xiz) : (-2.0f * xix);
  Areg[1] = hi ? (1.0f)        : (-2.0f * xiy);

  // Register-resident sorted top-K (ascending; [KNN-1] = current worst).
  float kd[KNN]; int ki[KNN];
#pragma unroll
  for (int q = 0; q < KNN; ++q) { kd[q] = FBIG; ki[q] = 0; }

  for (int t = 0; t < NTILES; ++t) {
    const int j0 = t * 16;
    const int jc = j0 + col;
    // prefetch 8 tiles ahead (x fits in L2; this keeps L0 warm)
    __builtin_prefetch(X + 3 * (jc + 16 * 8), 0, 1);

    const float xjx = X[3 * jc + 0];
    const float xjy = X[3 * jc + 1];
    const float xjz = X[3 * jc + 2];
    const float sqj = xjx * xjx + xjy * xjy + xjz * xjz;

    // B tile (4x16): VGPR0 = rows K0/K2, VGPR1 = rows K1/K3 by lane half.
    v2f Breg;
    Breg[0] = hi ? xjz : xjx;
    Breg[1] = hi ? sqj : xjy;

    v8f acc = {};
    acc = __builtin_amdgcn_wmma_f32_16x16x4_f32(
        /*neg_a=*/false, Areg, /*neg_b=*/false, Breg,
        /*c_mod=*/(short)0, acc, /*reuse_a=*/false, /*reuse_b=*/false);

    // C layout: VGPR r holds (M=r, N=col) for lanes 0-15, (M=r+8) for 16-31.
#pragma unroll
    for (int r = 0; r < 8; ++r)
      tileS[wave][r + hi * 8][col] = acc[r];

    // Same-wave LDS ops are in-order; lane reads its row's 8 columns.
#pragma unroll
    for (int c = 0; c < 8; ++c) {
      const int jj = j0 + hi * 8 + c;
      const float d = tileS[wave][col][hi * 8 + c];
      if (jj != myI && d < kd[KNN - 1]) {
        kd[KNN - 1] = d; ki[KNN - 1] = jj;
#pragma unroll
        for (int q = KNN - 1; q > 0; --q) {
          if (kd[q] < kd[q - 1]) {
            float td = kd[q - 1]; kd[q - 1] = kd[q]; kd[q] = td;
            int   ti = ki[q - 1]; ki[q - 1] = ki[q]; ki[q] = ti;
          }
        }
      }
    }
  }

  // ---- dump both sorted half-lists for this row, then merge on hi==0 lane
#pragma unroll
  for (int q = 0; q < KNN; ++q) {
    mD[wave][col][hi * KNN + q] = kd[q];
    mI[wave][col][hi * KNN + q] = ki[q];
  }
  __syncthreads();

  if (hi == 0) {
    float dxv[KNN], dyv[KNN], dzv[KNN];
    int ia = 0, ib = KNN;
#pragma unroll
    for (int m = 0; m < KNN; ++m) {
      const float da = (ia < KNN)     ? mD[wave][col][ia] : FBIG;
      const float db = (ib < 2 * KNN) ? mD[wave][col][ib] : FBIG;
      int j;
      if (da <= db) { j = mI[wave][col][ia]; ++ia; }
      else          { j = mI[wave][col][ib]; ++ib; }
      dxv[m] = X[3 * j + 0] - xix;
      dyv[m] = X[3 * j + 1] - xiy;
      dzv[m] = X[3 * j + 2] - xiz;
    }

    float* op = Out + (size_t)myI * COUT;
#pragma unroll 4
    for (int o = 0; o < COUT; ++o) {
      const float w0 = Wsh[6 * o + 0], w1 = Wsh[6 * o + 1], w2 = Wsh[6 * o + 2];
      const float w3 = Wsh[6 * o + 3], w4 = Wsh[6 * o + 4], w5 = Wsh[6 * o + 5];
      const float base = Bsh[o] + w0 * xix + w1 * xiy + w2 * xiz;
      float mo = -FBIG;
#pragma unroll
      for (int m = 0; m < KNN; ++m)
        mo = fmaxf(mo, base + w3 * dxv[m] + w4 * dyv[m] + w5 * dzv[m]);
      // relu(max) == max(relu) since relu is monotone
      op[o] = fmaxf(mo, 0.0f);
    }
  }
}

extern "C" void kernel_launch(void* const* d_in, const int* in_sizes, int n_in,
                              void* d_out, int out_size, void* d_ws, size_t ws_size,
                              hipStream_t stream) {
  (void)in_sizes; (void)n_in; (void)out_size; (void)d_ws; (void)ws_size;
  const float* X  = (const float*)d_in[0];  // [16384,3]
  const float* W  = (const float*)d_in[1];  // [32,6]
  const float* Bv = (const float*)d_in[2];  // [32]
  float* Out = (float*)d_out;               // [16384,32]

  dim3 grid(NPTS / ROWS_PER_BLOCK);         // 256 blocks
  dim3 block(BLOCK);                        // 128 threads = 4 waves
  edgeconv_knn_kernel<<<grid, block, 0, stream>>>(X, W, Bv, Out);
}